// MemN2N_16363825398453
// MI455X (gfx1250) — compile-verified
//
#include <hip/hip_runtime.h>
#include <hip/hip_bf16.h>

// Problem constants (match reference)
#define BB 32
#define MM 50
#define SS 20
#define VV 2000
#define EE 128
#define HOPS 3

typedef float v2f __attribute__((ext_vector_type(2)));
typedef float v8f __attribute__((ext_vector_type(8)));

// Workspace layout (bytes)
//   s_idx : int[B*M*S]  = 32000 ints   @ 0
//   q_idx : int[B*S]    = 640 ints     @ 128000
//   u_buf : float[B*E]  = 4096 floats  @ 131072
//   m_tab : float[4*B*M*E] = 819200 f  @ 147456   (t=0:A1, 1:C0, 2:C1, 3:C2)
#define OFF_SIDX 0
#define OFF_QIDX 128000
#define OFF_U    131072
#define OFF_MTAB 147456

// ---------------------------------------------------------------------------
// Kernel 1: one-hot -> index extraction. Memory-bound pass over 256 MB.
// Each row of V=2000 floats has exactly one 1.0, so one unguarded store/row.
// V % 4 == 0 so float4 vectors never straddle rows.
// ---------------------------------------------------------------------------
__global__ void onehot_to_idx(const float4* __restrict__ src,
                              int* __restrict__ idx_out,
                              long long n4) {
    long long i = (long long)blockIdx.x * blockDim.x + threadIdx.x;
    long long stride = (long long)gridDim.x * blockDim.x;
    for (; i < n4; i += stride) {
        if (i + stride < n4) {
            __builtin_prefetch(src + i + stride, 0, 0);  // global_prefetch_b8
        }
        float4 x = src[i];
        if (x.x != 0.f || x.y != 0.f || x.z != 0.f || x.w != 0.f) {
            int off = (x.x != 0.f) ? 0 : (x.y != 0.f) ? 1 : (x.z != 0.f) ? 2 : 3;
            long long e = i * 4 + off;
            int row = (int)(e / VV);
            int v   = (int)(e - (long long)row * VV);
            idx_out[row] = v;
        }
    }
}

// ---------------------------------------------------------------------------
// Kernel 2: m_tab[t][b][m][e] = sum_s enc[s][e] * table_t[idx][e]
// (idx==0 is the nil word -> contributes 0; tables are stored without nil row)
// One block per (b,m); 128 threads = one thread per e.
// ---------------------------------------------------------------------------
__global__ void build_m_tables(const int* __restrict__ s_idx,
                               const float* __restrict__ A1,       // (V-1,E)
                               const float* __restrict__ C,        // (3,V-1,E)
                               const float* __restrict__ enc,      // (S,E)
                               float* __restrict__ m_tab) {
    const int bm = blockIdx.x;      // 0..B*M-1
    const int e  = threadIdx.x;     // 0..127
    __shared__ int idx_s[SS];
    if (threadIdx.x < SS) idx_s[threadIdx.x] = s_idx[bm * SS + threadIdx.x];
    __syncthreads();

    #pragma unroll
    for (int t = 0; t < 4; ++t) {
        const float* tab = (t == 0) ? A1 : (C + (size_t)(t - 1) * (VV - 1) * EE);
        float acc = 0.f;
        #pragma unroll
        for (int s = 0; s < SS; ++s) {
            int id = idx_s[s];
            float val = (id != 0) ? tab[(size_t)(id - 1) * EE + e] : 0.f;
            acc = fmaf(enc[s * EE + e], val, acc);
        }
        m_tab[((size_t)t * BB * MM + bm) * EE + e] = acc;
    }
}

// ---------------------------------------------------------------------------
// Kernel 3: fused query embedding + 3 hops. One block per batch (128 thr).
// u lives in LDS; dotted/softmax over M=50 per hop.
// ---------------------------------------------------------------------------
__global__ void hops_fused(const int* __restrict__ q_idx,
                           const float* __restrict__ A1,
                           const float* __restrict__ enc,
                           const float* __restrict__ m_tab,
                           float* __restrict__ u_out) {
    const int b = blockIdx.x;
    const int e = threadIdx.x;            // 0..127
    __shared__ float u_s[EE];
    __shared__ float p_s[MM];
    __shared__ int   qi_s[SS];

    if (e < SS) qi_s[e] = q_idx[b * SS + e];
    __syncthreads();

    // u = sum_s enc[s]*A1[q_idx[b,s]]
    float u = 0.f;
    #pragma unroll
    for (int s = 0; s < SS; ++s) {
        int id = qi_s[s];
        float val = (id != 0) ? A1[(size_t)(id - 1) * EE + e] : 0.f;
        u = fmaf(enc[s * EE + e], val, u);
    }
    u_s[e] = u;
    __syncthreads();

    for (int h = 0; h < HOPS; ++h) {
        // dotted[m] = <m_A[b,m,:], u>
        if (e < MM) {
            const float* mA = m_tab + ((size_t)h * BB * MM + (size_t)b * MM + e) * EE;
            float d = 0.f;
            for (int k = 0; k < EE; ++k) d = fmaf(mA[k], u_s[k], d);
            p_s[e] = d;
        }
        __syncthreads();

        // softmax over M (redundant per-thread scalar reductions; M=50)
        float mx = p_s[0];
        for (int m = 1; m < MM; ++m) mx = fmaxf(mx, p_s[m]);
        __syncthreads();                 // all done reading raw logits
        if (e < MM) p_s[e] = __expf(p_s[e] - mx);
        __syncthreads();
        float sum = 0.f;
        for (int m = 0; m < MM; ++m) sum += p_s[m];
        float inv = 1.f / sum;

        // o_k[e] = sum_m probs[m] * m_C[b,m,e];  u += o_k
        const float* mC = m_tab + ((size_t)(h + 1) * BB * MM + (size_t)b * MM) * EE;
        float acc = 0.f;
        for (int m = 0; m < MM; ++m) acc = fmaf(p_s[m], mC[(size_t)m * EE + e], acc);
        u = fmaf(acc, inv, u);
        u_s[e] = u;
        __syncthreads();
    }
    u_out[(size_t)b * EE + e] = u;
}

// ---------------------------------------------------------------------------
// Kernel 4: out = u @ W.T via V_WMMA_F32_16X16X4_F32 (full fp32 precision).
// out is (B=32, V-1=1999), K = E = 128 -> 32 chained WMMAs per 16x16 tile.
// One wave per tile: grid (125, 2). A: 16x4 f32 frag = v2f per lane
// (lanes 0-15 hold K={k0,k0+1}, lanes 16-31 hold K={k0+2,k0+3}).
// C/D: v8f; lane half selects M 0-7 vs 8-15, N = lane & 15.
// ---------------------------------------------------------------------------
__global__ void out_gemm_wmma(const float* __restrict__ u,   // (B,E)
                              const float* __restrict__ W,   // (V-1,E)
                              float* __restrict__ out) {     // (B,V-1)
    const int l  = threadIdx.x;        // 0..31 (wave32)
    const int n0 = blockIdx.x * 16;    // output column tile
    const int b0 = blockIdx.y * 16;    // output row tile
    const int lane16 = l & 15;
    const int kh     = l >> 4;         // 0 or 1

    const int n  = n0 + lane16;
    const int nc = (n < (VV - 1)) ? n : (VV - 2);   // clamp ragged-edge loads

    const float* uRow = u + (size_t)(b0 + lane16) * EE;
    const float* wRow = W + (size_t)nc * EE;

    v8f c = {0.f, 0.f, 0.f, 0.f, 0.f, 0.f, 0.f, 0.f};
    #pragma unroll 4
    for (int k0 = 0; k0 < EE; k0 += 4) {
        const int ka = k0 + 2 * kh;
        v2f a = *(const v2f*)(uRow + ka);     // A[M=lane16][K=ka,ka+1]
        v2f bf = *(const v2f*)(wRow + ka);    // B[K=ka,ka+1][N=lane16] = W[n][k]
        c = __builtin_amdgcn_wmma_f32_16x16x4_f32(
                /*neg_a=*/false, a, /*neg_b=*/false, bf,
                /*c_mod=*/(short)0, c, /*reuse_a=*/false, /*reuse_b=*/false);
    }

    if (n < (VV - 1)) {
        #pragma unroll
        for (int r = 0; r < 8; ++r) {
            int row = b0 + r + 8 * kh;     // M = r + 8*(lane>=16)
            out[(size_t)row * (VV - 1) + n] = c[r];
        }
    }
}

// ---------------------------------------------------------------------------
extern "C" void kernel_launch(void* const* d_in, const int* in_sizes, int n_in,
                              void* d_out, int out_size, void* d_ws, size_t ws_size,
                              hipStream_t stream) {
    const float* stories = (const float*)d_in[0];   // (B,M,S,V)
    const float* queries = (const float*)d_in[1];   // (B,S,V)
    const float* A1      = (const float*)d_in[2];   // (V-1,E)
    const float* C       = (const float*)d_in[3];   // (3,V-1,E)
    const float* enc     = (const float*)d_in[4];   // (S,E)
    float*       out     = (float*)d_out;           // (B,V-1)

    char* ws = (char*)d_ws;
    int*   s_idx = (int*)(ws + OFF_SIDX);
    int*   q_idx = (int*)(ws + OFF_QIDX);
    float* u_buf = (float*)(ws + OFF_U);
    float* m_tab = (float*)(ws + OFF_MTAB);

    // 1) one-hot -> indices (the only HBM-heavy pass: 256 MB)
    {
        long long n4s = (long long)BB * MM * SS * VV / 4;   // 16M float4
        onehot_to_idx<<<8192, 256, 0, stream>>>((const float4*)stories, s_idx, n4s);
        long long n4q = (long long)BB * SS * VV / 4;        // 320K float4
        onehot_to_idx<<<1250, 256, 0, stream>>>((const float4*)queries, q_idx, n4q);
    }

    // 2) memory tables m[t][b][m][e]
    build_m_tables<<<BB * MM, EE, 0, stream>>>(s_idx, A1, C, enc, m_tab);

    // 3) fused query embedding + 3 hops
    hops_fused<<<BB, EE, 0, stream>>>(q_idx, A1, enc, m_tab, u_buf);

    // 4) output GEMM with fp32 WMMA: W = C_lookups[2]
    const float* W = C + (size_t)2 * (VV - 1) * EE;
    dim3 grid((VV - 1 + 15) / 16, BB / 16);   // (125, 2)
    out_gemm_wmma<<<grid, 32, 0, stream>>>(u_buf, W, out);
}